// Least_Squares_weight_65274912965339
// MI455X (gfx1250) — compile-verified
//
#include <hip/hip_runtime.h>
#include <hip/hip_bf16.h>

// ---------------------------------------------------------------------------
// Weighted 2-D Procrustes (Kabsch) — B=65536 batches, N=256 points.
// Memory-bound: 335.5 MB in / 0.75 MB out -> ~14.4 us floor @ 23.3 TB/s.
// 2x2 SVD replaced by closed form: theta = atan2(H01-H10, H00+H11).
// Pass 1: one wave32 per batch, CDNA5 async global->LDS B128 streaming
//         (ASYNCcnt), 14 moment sums, shfl_xor wave reduction, SoA store;
//         block-level partial of sum(w) (cuts pass-2 traffic 8x).
// Pass 2: global weight sum over 8192 block partials (single-block tree).
// Pass 3: one thread per batch epilogue.
// ---------------------------------------------------------------------------

#ifndef __has_builtin
#define __has_builtin(x) 0
#endif

#if __has_builtin(__builtin_amdgcn_global_load_async_to_lds_b128)
#define USE_ASYNC_LDS 1
#else
#define USE_ASYNC_LDS 0
#endif

#define NPTS 256      // points per batch
#define WPB  8        // waves (=batches) per 256-thread block

#if USE_ASYNC_LDS
// Probe-verified: param 1 of the async builtin is int4 in addrspace(1).
typedef int v4i_t __attribute__((ext_vector_type(4)));
typedef __attribute__((address_space(1))) v4i_t gv4i;   // global int4*
typedef __attribute__((address_space(3))) v4i_t lv4i;   // LDS int4*
#endif

// accumulate one point into the 14 moment sums
#define ACC_POINT(ax, ay, bx, by, wt)                                   \
  do {                                                                  \
    const float _w = (wt);                                              \
    const float _ax = (ax), _ay = (ay), _bx = (bx), _by = (by);         \
    const float _w2 = _w * _w;                                          \
    sw  += _w;                                                          \
    sw2 += _w2;                                                         \
    swAx = fmaf(_w, _ax, swAx);  swAy = fmaf(_w, _ay, swAy);            \
    swBx = fmaf(_w, _bx, swBx);  swBy = fmaf(_w, _by, swBy);            \
    const float _wax = _w2 * _ax, _way = _w2 * _ay;                     \
    qx += _wax;  qy += _way;                                            \
    tx = fmaf(_w2, _bx, tx);  ty = fmaf(_w2, _by, ty);                  \
    sxx = fmaf(_wax, _bx, sxx);  sxy = fmaf(_wax, _by, sxy);            \
    syx = fmaf(_way, _bx, syx);  syy = fmaf(_way, _by, syy);            \
  } while (0)

__global__ __launch_bounds__(256) void k_batch_reduce(
    const float* __restrict__ pstA, const float* __restrict__ pstB,
    const float* __restrict__ wgt, float* __restrict__ sums,
    float* __restrict__ swPart, int B) {
  const int lane = threadIdx.x & 31;
  const int wave = threadIdx.x >> 5;
  const int b = blockIdx.x * WPB + wave;
  const bool active = (b < B);

#if USE_ASYNC_LDS
  __shared__ __align__(16) float lds[WPB][1280];  // per wave: A512 B512 W256
#endif
  __shared__ float swRed[WPB];

  float swAx = 0.f, swAy = 0.f, swBx = 0.f, swBy = 0.f;
  float qx = 0.f, qy = 0.f, tx = 0.f, ty = 0.f;            // sum w^2 * a / b
  float sxx = 0.f, sxy = 0.f, syx = 0.f, syy = 0.f;        // sum w^2 a_c b_d
  float sw2 = 0.f, sw = 0.f;

  if (active) {  // wave-uniform branch
    const size_t baseP = (size_t)b * (NPTS * 2);
    const size_t baseW = (size_t)b * NPTS;

#if USE_ASYNC_LDS
    // CDNA5 async global->LDS streaming (ASYNCcnt-tracked). Each wave copies
    // only its own 5 KB batch slice: no barrier needed, just s_wait_asynccnt.
    float* wb = lds[wave];
#pragma unroll
    for (int k = 0; k < 4; ++k) {
      const int idx = k * 32 + lane;
      __builtin_amdgcn_global_load_async_to_lds_b128(
          (gv4i*)(pstA + baseP + (size_t)idx * 4), (lv4i*)(wb + idx * 4), 0, 0);
      __builtin_amdgcn_global_load_async_to_lds_b128(
          (gv4i*)(pstB + baseP + (size_t)idx * 4), (lv4i*)(wb + 512 + idx * 4),
          0, 0);
    }
#pragma unroll
    for (int j = 0; j < 2; ++j) {
      const int idx = j * 32 + lane;
      __builtin_amdgcn_global_load_async_to_lds_b128(
          (gv4i*)(wgt + baseW + (size_t)idx * 4), (lv4i*)(wb + 1024 + idx * 4),
          0, 0);
    }
#if __has_builtin(__builtin_amdgcn_s_wait_asynccnt)
    __builtin_amdgcn_s_wait_asynccnt(0);
#else
    asm volatile("s_wait_asynccnt 0" ::: "memory");
#endif
#else
    __builtin_prefetch(pstA + baseP, 0, 0);
    __builtin_prefetch(pstB + baseP, 0, 0);
    __builtin_prefetch(wgt + baseW, 0, 0);
#endif

#pragma unroll
    for (int k = 0; k < 4; ++k) {
      const int idx = k * 32 + lane;  // covers points 2*idx, 2*idx+1
#if USE_ASYNC_LDS
      const float4 a4 = ((const float4*)wb)[idx];
      const float4 b4 = ((const float4*)(wb + 512))[idx];
      const float2 w2 = ((const float2*)(wb + 1024))[idx];
#else
      const float4 a4 = *(const float4*)(pstA + baseP + (size_t)idx * 4);
      const float4 b4 = *(const float4*)(pstB + baseP + (size_t)idx * 4);
      const float2 w2 = *(const float2*)(wgt + baseW + (size_t)idx * 2);
#endif
      ACC_POINT(a4.x, a4.y, b4.x, b4.y, w2.x);
      ACC_POINT(a4.z, a4.w, b4.z, b4.w, w2.y);
    }
  }

  float acc[14] = {swAx, swAy, swBx, swBy, qx,  qy,  tx, ty,
                   sxx,  sxy,  syx,  syy,  sw2, sw};
#pragma unroll
  for (int i = 0; i < 14; ++i) {
    float v = acc[i];
#pragma unroll
    for (int off = 16; off > 0; off >>= 1) v += __shfl_xor(v, off, 32);
    acc[i] = v;
  }

  if (lane == 0) {
    if (active) {
#pragma unroll
      for (int i = 0; i < 13; ++i) sums[(size_t)i * B + b] = acc[i];
    }
    swRed[wave] = active ? acc[13] : 0.f;  // per-wave sum(w)
  }
  __syncthreads();
  if (threadIdx.x == 0) {
    float s = 0.f;
#pragma unroll
    for (int i = 0; i < WPB; ++i) s += swRed[i];
    swPart[blockIdx.x] = s;  // one partial per block (8192 total)
  }
}

// deterministic global sum of block partials -> *out
__global__ __launch_bounds__(1024) void k_global_ws(
    const float* __restrict__ part, float* __restrict__ out, int n) {
  __shared__ float red[1024];
  float s = 0.f;
  for (int i = threadIdx.x; i < n; i += 1024) s += part[i];
  red[threadIdx.x] = s;
  __syncthreads();
  for (int stride = 512; stride > 0; stride >>= 1) {
    if ((int)threadIdx.x < stride) red[threadIdx.x] += red[threadIdx.x + stride];
    __syncthreads();
  }
  if (threadIdx.x == 0) out[0] = red[0];
}

__global__ __launch_bounds__(256) void k_finalize(
    const float* __restrict__ sums, const float* __restrict__ wsPtr,
    float* __restrict__ out, int B) {
  const int b = blockIdx.x * 256 + threadIdx.x;
  if (b >= B) return;
  const float inv = 1.0f / wsPtr[0];

  const float swAx = sums[(size_t)0 * B + b], swAy = sums[(size_t)1 * B + b];
  const float swBx = sums[(size_t)2 * B + b], swBy = sums[(size_t)3 * B + b];
  const float qx = sums[(size_t)4 * B + b], qy = sums[(size_t)5 * B + b];
  const float tx = sums[(size_t)6 * B + b], ty = sums[(size_t)7 * B + b];
  const float sxx = sums[(size_t)8 * B + b], sxy = sums[(size_t)9 * B + b];
  const float syx = sums[(size_t)10 * B + b], syy = sums[(size_t)11 * B + b];
  const float sw2 = sums[(size_t)12 * B + b];

  const float GAx = swAx * inv, GAy = swAy * inv;
  const float GBx = swBx * inv, GBy = swBy * inv;

  // H_cd = sum w^2 (a_c-GA_c)(b_d-GB_d), expanded in raw moments
  const float H00 = sxx - GAx * tx - GBx * qx + GAx * GBx * sw2;
  const float H01 = sxy - GAx * ty - GBy * qx + GAx * GBy * sw2;
  const float H10 = syx - GAy * tx - GBx * qy + GAy * GBx * sw2;
  const float H11 = syy - GAy * ty - GBy * qy + GAy * GBy * sw2;

  // det-corrected Kabsch rotation == argmax over SO(2) of tr(R*H)
  const float alpha = H00 + H11;     // cos component
  const float beta = H01 - H10;      // sin component
  const float r = sqrtf(alpha * alpha + beta * beta);
  const float c = alpha / r;
  const float s = beta / r;

  // mirror reference: acos(clip(tr/2)) * R10 / |R01|   (NaN when s == 0)
  const float cc = fminf(fmaxf(c, -1.0f), 1.0f);
  const float theta = acosf(cc) * s / fabsf(s);

  const float t0 = GBx - (c * GAx - s * GAy);
  const float t1 = GBy - (s * GAx + c * GAy);

  out[b] = theta;
  out[(size_t)B + b] = t0;
  out[(size_t)2 * B + b] = t1;
}

extern "C" void kernel_launch(void* const* d_in, const int* in_sizes, int n_in,
                              void* d_out, int out_size, void* d_ws,
                              size_t ws_size, hipStream_t stream) {
  const float* pstA = (const float*)d_in[0];
  const float* pstB = (const float*)d_in[1];
  const float* wgt = (const float*)d_in[2];
  float* out = (float*)d_out;
  const int B = in_sizes[2] / NPTS;  // weight is [B, NPTS, 1]
  const int nBlocks = (B + WPB - 1) / WPB;

  // workspace layout: 13 arrays of B floats | nBlocks partials | ws scalar
  float* sums = (float*)d_ws;
  float* swPart = sums + (size_t)13 * B;
  float* wsScalar = swPart + nBlocks;

  k_batch_reduce<<<dim3(nBlocks), dim3(256), 0, stream>>>(pstA, pstB, wgt,
                                                          sums, swPart, B);
  k_global_ws<<<dim3(1), dim3(1024), 0, stream>>>(swPart, wsScalar, nBlocks);
  k_finalize<<<dim3((B + 255) / 256), dim3(256), 0, stream>>>(sums, wsScalar,
                                                              out, B);
}